// SpectralLocalUnit_51445118272193
// MI455X (gfx1250) — compile-verified
//
#include <hip/hip_runtime.h>
#include <hip/hip_bf16.h>

typedef float v2f __attribute__((ext_vector_type(2)));
typedef float v8f __attribute__((ext_vector_type(8)));

#define BSZ 32
#define CSZ 256
#define HSZ 96
#define WSZ 96
#define HW  (HSZ * WSZ)   // 9216

// ---------------------------------------------------------------------------
// Kernel 1: global average pool over H,W.  One block per (b,c) plane.
// 9216 floats = 2304 float4; 256 threads x 9 float4 each, then LDS tree.
// ---------------------------------------------------------------------------
__global__ void slu_pool_kernel(const float* __restrict__ x,
                                float* __restrict__ pooled) {
    __shared__ float red[256];
    const int plane = blockIdx.x;  // b*C + c
    const float4* p4 = reinterpret_cast<const float4*>(x + (size_t)plane * HW);
    float s = 0.f;
#pragma unroll
    for (int i = 0; i < 9; ++i) {
        float4 v = p4[threadIdx.x + i * 256];
        s += v.x + v.y + v.z + v.w;
    }
    red[threadIdx.x] = s;
    __syncthreads();
    for (int off = 128; off > 0; off >>= 1) {
        if (threadIdx.x < (unsigned)off) red[threadIdx.x] += red[threadIdx.x + off];
        __syncthreads();
    }
    if (threadIdx.x == 0) pooled[plane] = red[0] * (1.0f / HW);
}

// ---------------------------------------------------------------------------
// Kernels 2a/2b: D = A(MxK) @ Wt(NxK)^T + bias, optional exact GELU.
// One wave (32 threads) per 16x16 output tile, K stepped by 4 with
// V_WMMA_F32_16X16X4_F32 (fp32 matrix core -> exact math).
//
// A-tile VGPR layout (16x4 f32): lane L (L<16): row=L, holds K=k0,k0+1;
// lane L (L>=16): row=L-16, holds K=k0+2,k0+3.  B-tile (4x16) mirrors this
// per column.  Both collapse to one aligned float2 load at
// koff = k0 + 2*(lane>>4).
// C/D (16x16 f32, 8 VGPRs): VGPR r -> row r+8*(lane>>4), col lane&15.
// EXEC is all ones (blockDim=32, no divergence before WMMA), per ISA §7.12.
// ---------------------------------------------------------------------------
__global__ void slu_wmma_gemm_bias(const float* __restrict__ A,    // M x K
                                   const float* __restrict__ Wt,   // N x K
                                   const float* __restrict__ bias, // N
                                   float* __restrict__ D,          // M x N
                                   int N, int K, int apply_gelu) {
    const int lane   = threadIdx.x;        // 0..31
    const int ntiles = N >> 4;
    const int mtile  = blockIdx.x / ntiles;
    const int ntile  = blockIdx.x % ntiles;
    const int half   = lane >> 4;          // 0 | 1
    const int l15    = lane & 15;
    const int arow   = (mtile << 4) + l15; // A row handled by this lane
    const int bcol   = (ntile << 4) + l15; // output column n

    v8f acc = {};
    for (int k0 = 0; k0 < K; k0 += 4) {
        const int koff = k0 + (half << 1);
        v2f a = *reinterpret_cast<const v2f*>(A  + (size_t)arow * K + koff);
        v2f b = *reinterpret_cast<const v2f*>(Wt + (size_t)bcol * K + koff);
        acc = __builtin_amdgcn_wmma_f32_16x16x4_f32(
            /*neg_a=*/false, a, /*neg_b=*/false, b,
            /*c_mod=*/(short)0, acc, /*reuse_a=*/false, /*reuse_b=*/false);
    }

    const float bv = bias[bcol];
#pragma unroll
    for (int r = 0; r < 8; ++r) {
        const int grow = (mtile << 4) + r + (half << 3);
        float u = acc[r] + bv;
        if (apply_gelu)  // exact GELU: 0.5*u*(1+erf(u/sqrt(2)))
            u = 0.5f * u * (1.0f + erff(u * 0.70710678118654752440f));
        D[(size_t)grow * N + bcol] = u;
    }
}

// ---------------------------------------------------------------------------
// Kernel 3: per-(b,c) dynamic 3x3 stencil with reflect padding.
// One block per (b,c) plane, 96 threads = one column each.  Rolling 3x3
// register window: 3 new coalesced loads per output pixel instead of 9.
// Reflect: row/col -1 -> 1, 96 -> 94.  At y=95 the bottom row (reflect(96)=94)
// is exactly the post-shift top row, so reuse registers (no reload).
// ---------------------------------------------------------------------------
__global__ void slu_dynconv_kernel(const float* __restrict__ x,
                                   const float* __restrict__ wdyn,
                                   float* __restrict__ out) {
    const int plane = blockIdx.x;   // b*C + c
    const int tx    = threadIdx.x;  // 0..95 (column)
    const float* p  = x   + (size_t)plane * HW;
    float*       o  = out + (size_t)plane * HW;
    const float* wp = wdyn + (size_t)plane * 9;

    const float w0 = wp[0], w1 = wp[1], w2 = wp[2];
    const float w3 = wp[3], w4 = wp[4], w5 = wp[5];
    const float w6 = wp[6], w7 = wp[7], w8 = wp[8];

    const int xl = (tx == 0)       ? 1       : tx - 1;
    const int xr = (tx == WSZ - 1) ? WSZ - 2 : tx + 1;

    float l0, m0, r0, l1, m1, r1, l2, m2, r2;
    // center row (y=0)
    l1 = p[xl];       m1 = p[tx];       r1 = p[xr];
    // bottom row (y=1)
    l2 = p[WSZ + xl]; m2 = p[WSZ + tx]; r2 = p[WSZ + xr];
    // reflected row -1 == row 1
    l0 = l2; m0 = m2; r0 = r2;

    for (int y = 0; y < HSZ; ++y) {
        float acc = w0 * l0;
        acc = fmaf(w1, m0, acc);
        acc = fmaf(w2, r0, acc);
        acc = fmaf(w3, l1, acc);
        acc = fmaf(w4, m1, acc);
        acc = fmaf(w5, r1, acc);
        acc = fmaf(w6, l2, acc);
        acc = fmaf(w7, m2, acc);
        acc = fmaf(w8, r2, acc);
        o[y * WSZ + tx] = acc;

        if (y == HSZ - 1) break;
        l0 = l1; m0 = m1; r0 = r1;
        l1 = l2; m1 = m2; r1 = r2;
        if (y + 2 < HSZ) {
            const int row = (y + 2) * WSZ;
            l2 = p[row + xl]; m2 = p[row + tx]; r2 = p[row + xr];
        } else {  // reflect(96)=94 == new top row
            l2 = l0; m2 = m0; r2 = r0;
        }
    }
}

// ---------------------------------------------------------------------------
// Launch: pool -> WMMA gemm1(+GELU) -> WMMA gemm2 -> stencil, all on stream.
// Workspace layout (floats): pooled[8192] | h[8192] | wdyn[73728]  (~360 KB)
// ---------------------------------------------------------------------------
extern "C" void kernel_launch(void* const* d_in, const int* in_sizes, int n_in,
                              void* d_out, int out_size, void* d_ws, size_t ws_size,
                              hipStream_t stream) {
    const float* x  = (const float*)d_in[0];
    const float* w1 = (const float*)d_in[1];
    const float* b1 = (const float*)d_in[2];
    const float* w2 = (const float*)d_in[3];
    const float* b2 = (const float*)d_in[4];
    float* out = (float*)d_out;

    float* ws     = (float*)d_ws;
    float* pooled = ws;                    // 32*256
    float* h      = ws + BSZ * CSZ;        // 32*256
    float* wdyn   = ws + 2 * BSZ * CSZ;    // 32*2304

    // 1) global average pool
    slu_pool_kernel<<<BSZ * CSZ, 256, 0, stream>>>(x, pooled);

    // 2a) h = gelu(pooled @ w1^T + b1): M=32,N=256,K=256 -> 2*16 tiles
    slu_wmma_gemm_bias<<<(BSZ / 16) * (CSZ / 16), 32, 0, stream>>>(
        pooled, w1, b1, h, CSZ, CSZ, 1);

    // 2b) wdyn = h @ w2^T + b2: M=32,N=2304,K=256 -> 2*144 tiles
    slu_wmma_gemm_bias<<<(BSZ / 16) * ((CSZ * 9) / 16), 32, 0, stream>>>(
        h, w2, b2, wdyn, CSZ * 9, CSZ, 0);

    // 3) dynamic depthwise 3x3 with reflect pad
    slu_dynconv_kernel<<<BSZ * CSZ, WSZ, 0, stream>>>(x, wdyn, out);
}